// deepLPM_31903017074980
// MI455X (gfx1250) — compile-verified
//
#include <hip/hip_runtime.h>
#include <hip/hip_bf16.h>

typedef float v2f __attribute__((ext_vector_type(2)));
typedef float v8f __attribute__((ext_vector_type(8)));

#define DET 1e-16f

// DPP16 butterfly add over each 16-lane group (all lanes end with group sum).
// ctrl: 0xB1=quad_perm xor1, 0x4E=quad_perm xor2, 0x141=row_half_mirror, 0x140=row_mirror
#define DPP_ADD(v, ctrl)                                                      \
  ((v) + __int_as_float(__builtin_amdgcn_update_dpp(                          \
             0, __float_as_int(v), (ctrl), 0xF, 0xF, true)))

// ---------------------------------------------------------------------------
// ws layout (floats): [0..15] N_k  [16..31] A_k  [32..287] S[k][p]
// ---------------------------------------------------------------------------

__global__ void em_init_kernel(float* __restrict__ acc) {
  int i = threadIdx.x;
  if (i < 288) acc[i] = 0.0f;
}

__launch_bounds__(256)
__global__ void em_main_kernel(const float* __restrict__ mu_phi,
                               const float* __restrict__ log_cov_phi,
                               const float* __restrict__ pi_k,
                               const float* __restrict__ mu_k,
                               const float* __restrict__ log_cov_k,
                               float* __restrict__ gamma_out,
                               float* __restrict__ acc,
                               int ntiles) {
  // per-wave 16x18 (padded, conflict-free) gamma tile + block reduction buf
  __shared__ float sG[8 * 288];
  __shared__ float red[288];

  const int tid  = threadIdx.x;
  const int lane = tid & 31;
  const int kk   = lane & 15;        // col index (k for GEMM1 D, p for GEMM2 D)
  const int hi   = (lane >> 4) & 1;  // lane-half select
  float* myG = &sG[(tid >> 5) * 288];

  for (int i = tid; i < 288; i += 256) red[i] = 0.0f;
  __syncthreads();

  // ---- per-k constants (k = kk), loaded once ----
  // mu_k in WMMA-B layout: slice s, V0 -> p=4s+2hi, V1 -> p=4s+1+2hi
  v2f bk[4];
#pragma unroll
  for (int s = 0; s < 4; ++s)
    bk[s] = *(const v2f*)(mu_k + kk * 16 + 4 * s + 2 * hi);
  float skp = 0.0f;
#pragma unroll
  for (int s = 0; s < 4; ++s) skp += bk[s].x * bk[s].x + bk[s].y * bk[s].y;
  const float skk  = skp + __shfl_xor(skp, 16, 32);  // ||mu_k||^2
  const float lpik = __logf(pi_k[kk]);
  const float lckv = log_cov_k[kk];
  const float ick  = __expf(-lckv);
  // logit[m,k] = kconst + ick*(dot + lcp[m]*8*e^lck - 0.5*u[m]),  u = e^lcp+||x||^2
  const float kconst = lpik - 8.0f * lckv + 8.0f - 0.5f * skk * ick;
  // 5th (augmentation) WMMA slice, B operand: K0 = 8*e^lck, K1 = -0.5
  v2f b5;
  b5.x = hi ? 0.0f : 8.0f * __expf(lckv);
  b5.y = hi ? 0.0f : -0.5f;

  const int gwave  = (blockIdx.x * 256 + tid) >> 5;
  const int nwaves = (gridDim.x * 256) >> 5;

  float nk_acc = 0.0f, ak_acc = 0.0f;
  v8f c2 = {0.f, 0.f, 0.f, 0.f, 0.f, 0.f, 0.f, 0.f};  // S[k][p] accumulator

  for (int t = gwave; t < ntiles; t += nwaves) {
    const int base = t * 16;
    const float* X = mu_phi + (size_t)base * 16;

    // speculative prefetch of next tile for this wave
    if (t + nwaves < ntiles) {
      const float* Xn = mu_phi + (size_t)(t + nwaves) * 16 * 16;
      if (lane < 8) __builtin_prefetch(Xn + lane * 32, 0, 1);
    }

    // ---- GEMM1 (+aug): dot[m][k] via 5x V_WMMA_F32_16X16X4_F32 ----
    v2f ax[4];
#pragma unroll
    for (int s = 0; s < 4; ++s)
      ax[s] = *(const v2f*)(X + kk * 16 + 4 * s + 2 * hi);  // A layout: row kk

    // per-row scalars (row = kk in "row layout"; only lanes 0-15 feed A5)
    float sp = 0.0f;
#pragma unroll
    for (int s = 0; s < 4; ++s) sp += ax[s].x * ax[s].x + ax[s].y * ax[s].y;
    const float ssq_row = sp + __shfl_xor(sp, 16, 32);  // ||x_m||^2
    const float lcp_row = log_cov_phi[base + kk];
    const float u_row   = __expf(lcp_row) + ssq_row;
    v2f a5;
    a5.x = hi ? 0.0f : lcp_row;
    a5.y = hi ? 0.0f : u_row;

    v8f c1 = {0.f, 0.f, 0.f, 0.f, 0.f, 0.f, 0.f, 0.f};
#pragma unroll
    for (int s = 0; s < 4; ++s)
      c1 = __builtin_amdgcn_wmma_f32_16x16x4_f32(false, ax[s], false, bk[s],
                                                 (short)0, c1, false, false);
    c1 = __builtin_amdgcn_wmma_f32_16x16x4_f32(false, a5, false, b5,
                                               (short)0, c1, false, false);

    // u[m] in C layout (needed for A_k accumulation)
    float uc[8];
#pragma unroll
    for (int r = 0; r < 8; ++r) uc[r] = __shfl(u_row, r + 8 * hi, 32);

    // ---- softmax over k, no max-sub (logits bounded, f32-safe) ----
    float g[8];
#pragma unroll
    for (int r = 0; r < 8; ++r) {
      const float e = __expf(fmaf(c1[r], ick, kconst));
      float sm = e;
      sm = DPP_ADD(sm, 0xB1);   // xor1
      sm = DPP_ADD(sm, 0x4E);   // xor2
      sm = DPP_ADD(sm, 0x141);  // row_half_mirror (xor-ish 4..7)
      sm = DPP_ADD(sm, 0x140);  // row_mirror (cross 8-boundary)
      g[r] = __fdividef(e, sm) + DET;
    }

    // ---- write gamma (streaming), accumulate N_k / A_k, stage G into LDS ----
#pragma unroll
    for (int r = 0; r < 8; ++r) {
      const int m = r + 8 * hi;
      __builtin_nontemporal_store(g[r], &gamma_out[(size_t)(base + m) * 16 + kk]);
      nk_acc += g[r];
      ak_acc += g[r] * uc[r];
      myG[m * 18 + kk] = g[r];
    }
    asm volatile("" ::: "memory");  // keep DS store->load order (same wave, in-order)

    // ---- GEMM2: S += G^T * X  (A from LDS transposed read, B coalesced global)
#pragma unroll
    for (int s = 0; s < 4; ++s) {
      v2f a2, b2;
      a2.x = myG[(4 * s + 2 * hi) * 18 + kk];
      a2.y = myG[(4 * s + 1 + 2 * hi) * 18 + kk];
      b2.x = X[(4 * s + 2 * hi) * 16 + kk];
      b2.y = X[(4 * s + 1 + 2 * hi) * 16 + kk];
      c2 = __builtin_amdgcn_wmma_f32_16x16x4_f32(false, a2, false, b2,
                                                 (short)0, c2, false, false);
    }
  }

  // ---- block-level reduction, then one set of global atomics per block ----
  atomicAdd(&red[kk], nk_acc);
  atomicAdd(&red[16 + kk], ak_acc);
#pragma unroll
  for (int r = 0; r < 8; ++r)
    atomicAdd(&red[32 + (r + 8 * hi) * 16 + kk], c2[r]);
  __syncthreads();
  for (int i = tid; i < 288; i += 256)
    unsafeAtomicAdd(&acc[i], red[i]);
}

__global__ void em_finalize_kernel(const float* __restrict__ acc,
                                   float* __restrict__ pi_new,
                                   float* __restrict__ mu_new,
                                   float* __restrict__ lc_new, int n) {
  const int k = threadIdx.x;
  if (k < 16) {
    const float nk = acc[k];
    const float ak = acc[16 + k];
    pi_new[k] = nk / (float)n;
    float msq = 0.0f;
#pragma unroll
    for (int p = 0; p < 16; ++p) {
      const float m = acc[32 + k * 16 + p] / nk;
      mu_new[k * 16 + p] = m;
      msq += m * m;
    }
    const float cov = (ak - msq * nk) / (16.0f * nk);
    lc_new[k] = logf(cov);
  }
}

extern "C" void kernel_launch(void* const* d_in, const int* in_sizes, int n_in,
                              void* d_out, int out_size, void* d_ws, size_t ws_size,
                              hipStream_t stream) {
  const float* mu_phi      = (const float*)d_in[0];
  const float* log_cov_phi = (const float*)d_in[1];
  const float* pi_k        = (const float*)d_in[2];
  const float* mu_k        = (const float*)d_in[3];
  const float* log_cov_k   = (const float*)d_in[4];

  const int n      = in_sizes[0] / 16;  // N (divisible by 16 here: 1,000,000)
  const int ntiles = n / 16;

  float* acc   = (float*)d_ws;
  float* gamma = (float*)d_out;
  float* pi_n  = gamma + (size_t)n * 16;
  float* mu_n  = pi_n + 16;
  float* lc_n  = mu_n + 256;

  em_init_kernel<<<1, 288, 0, stream>>>(acc);

  int blocks = 1024;                       // 8 waves/block -> 8192 waves
  int maxb   = (ntiles + 7) / 8;
  if (blocks > maxb) blocks = maxb;
  if (blocks < 1) blocks = 1;
  em_main_kernel<<<blocks, 256, 0, stream>>>(mu_phi, log_cov_phi, pi_k, mu_k,
                                             log_cov_k, gamma, acc, ntiles);

  em_finalize_kernel<<<1, 32, 0, stream>>>(acc, pi_n, mu_n, lc_n, n);
}